// ManifoldMessagePassing_50448685859294
// MI455X (gfx1250) — compile-verified
//
#include <hip/hip_runtime.h>
#include <cmath>

typedef __attribute__((ext_vector_type(2))) float v2f;
typedef __attribute__((ext_vector_type(8))) float v8f;
typedef __attribute__((ext_vector_type(4))) int   v4i;

#define NN 2048
#define DD 128
#define TILE 16
#define EPSF 1e-7f
#define MAXNORM (1.0f - 1e-5f)

#if __has_builtin(__builtin_amdgcn_global_load_async_to_lds_b128) && \
    __has_builtin(__builtin_amdgcn_s_wait_asynccnt)
#define USE_ASYNC 1
typedef __attribute__((address_space(1))) v4i gv4i;   // global int4 vector
typedef __attribute__((address_space(3))) v4i lv4i;   // LDS int4 vector
#else
#define USE_ASYNC 0
#endif

// ---------------------------------------------------------------------------
// Kernel 1: per-row precompute  px[i] = |x_i|^2 ,  invdeg[i] = 1/max(deg,1e-8)
// ---------------------------------------------------------------------------
__global__ __launch_bounds__(256)
void precompute_kernel(const float* __restrict__ x, const float* __restrict__ adj,
                       float* __restrict__ px, float* __restrict__ invdeg) {
    __shared__ float red[256];
    const int i = blockIdx.x;
    const int t = threadIdx.x;

    float s = 0.f;
    for (int j = t; j < NN; j += 256) s += adj[(size_t)i * NN + j];
    red[t] = s; __syncthreads();
    for (int off = 128; off > 0; off >>= 1) {
        if (t < off) red[t] += red[t + off];
        __syncthreads();
    }
    if (t == 0) invdeg[i] = 1.0f / fmaxf(red[0], 1e-8f);
    __syncthreads();

    float sx = 0.f;
    if (t < DD) { float v = x[(size_t)i * DD + t]; sx = v * v; }
    red[t] = sx; __syncthreads();
    for (int off = 128; off > 0; off >>= 1) {
        if (t < off) red[t] += red[t + off];
        __syncthreads();
    }
    if (t == 0) px[i] = red[0];
}

// ---------------------------------------------------------------------------
// 8 KB tile fills: 16 lanes x 128 cols of f32 (X_j tile) or 128 x 16 (W panel)
// One B128 transfer per lane per step; async path uses the gfx1250
// global_load_async_to_lds datapath (tracked on ASYNCcnt).
// ---------------------------------------------------------------------------
__device__ __forceinline__
void async_copy16(const float* __restrict__ gsrc_base, float* lbuf,
                  const int* goff, const int* loff) {
#pragma unroll
    for (int t = 0; t < 16; ++t) {
#if USE_ASYNC
        __builtin_amdgcn_global_load_async_to_lds_b128(
            (gv4i*)(gsrc_base + goff[t]),
            (lv4i*)(lbuf + loff[t]), 0, 0);
#else
        *(float4*)(lbuf + loff[t]) = *(const float4*)(gsrc_base + goff[t]);
#endif
    }
}

__device__ __forceinline__ void wait_tile_ready(bool more_in_flight) {
#if USE_ASYNC
    if (more_in_flight) __builtin_amdgcn_s_wait_asynccnt(16);
    else                __builtin_amdgcn_s_wait_asynccnt(0);
#else
    (void)more_in_flight;
    __syncthreads();   // single-wave workgroup: forces loadcnt/dscnt drain
#endif
}

// ---------------------------------------------------------------------------
// Kernel 2: fused manifold message passing. One wave (32 lanes) per 16 rows.
// ---------------------------------------------------------------------------
__global__ __launch_bounds__(32)
void manifold_mp_kernel(const float* __restrict__ x, const float* __restrict__ adj,
                        const float* __restrict__ W, const float* __restrict__ bias,
                        const float* __restrict__ px, const float* __restrict__ invdeg,
                        float* __restrict__ out) {
    __shared__ float ldsXJ[2 * TILE * DD];  // 16 KB: double-buffered X_j tile / W panel
    __shared__ float ldsG[TILE * TILE];     // C-layout -> A-layout transpose of G tile
    __shared__ float ldsMT[TILE * DD];      // 16x128 tangent mean for the weight GEMM

    const int lane = threadIdx.x;
    const int g = lane >> 4;     // half-wave group (C layout: M offset 8*g)
    const int l = lane & 15;     // position within group (N / M index)
    const int i0 = blockIdx.x * TILE;

    // Per-lane transfer index tables for the two 8KB tile shapes.
    int xg[16], xl[16];   // X_j tile: row-major 16x128
    int wg[16], wl[16];   // W panel: rows 0..127, 16 cols
#pragma unroll
    for (int t = 0; t < 16; ++t) {
        int flat = lane * 64 + t * 4;           // 0..2047
        xg[t] = ((flat >> 7) * DD) + (flat & 127);
        xl[t] = flat;
        int row = lane * 4 + (t >> 2);          // 0..127
        int seg = (t & 3) * 4;                  // 0,4,8,12
        wg[t] = row * DD + seg;                 // + nc*16 added at call site
        wl[t] = row * TILE + seg;
    }

    // Loop-invariant per-row values for the 8 M-rows this lane touches.
    float p_r[8], w_r[8];
#pragma unroll
    for (int r = 0; r < 8; ++r) {
        int gi = i0 + r + 8 * g;
        p_r[r] = px[gi];
        w_r[r] = invdeg[gi];
    }

    v8f acc[8] = {};      // mean_t tile accumulator, 16x128 in WMMA C layout
    float cacc[8];        // coefficient on x_i, partial over this lane's j's
#pragma unroll
    for (int r = 0; r < 8; ++r) cacc[r] = 0.f;

    const float* xiA = x + (size_t)(i0 + l) * DD + 2 * g;   // A-operand base (X_i)

    // Prologue: start DMA of the first X_j tile into buffer 0.
    async_copy16(x, &ldsXJ[0], xg, xl);

    for (int j0 = 0; j0 < NN; j0 += TILE) {
        const int buf = (j0 >> 4) & 1;
        float* cur = &ldsXJ[buf * (TILE * DD)];
        const bool has_next = (j0 + TILE) < NN;

        // Kick off the next tile's DMA into the other buffer, then wait for ours.
        if (has_next) {
            async_copy16(x + (size_t)(j0 + TILE) * DD, &ldsXJ[(buf ^ 1) * (TILE * DD)],
                         xg, xl);
            __builtin_prefetch(&adj[(size_t)(i0 + l) * NN + j0 + TILE], 0, 0);
        }
        wait_tile_ready(has_next);

        // ---- Stage 1: Gram tile U = X_i (16x128) * X_j^T  (32x v_wmma f32) --
        v8f U = {};
#pragma unroll
        for (int kc = 0; kc < 32; ++kc) {
            v2f a = *(const v2f*)(xiA + 4 * kc);                  // registers (hoisted)
            v2f b = *(const v2f*)(&cur[l * DD + 4 * kc + 2 * g]); // ds_load_b64
            U = __builtin_amdgcn_wmma_f32_16x16x4_f32(false, a, false, b,
                                                      (short)0, U, false, false);
        }

        // ---- per-pair scalar math: logmap coefficients ----
        float q = px[j0 + l];
#pragma unroll
        for (int r = 0; r < 8; ++r) {
            int mrow = r + 8 * g;
            float p = p_r[r];
            float u = U[r];
            float w = adj[(size_t)(i0 + mrow) * NN + j0 + l] * w_r[r];
            float den = fmaxf(1.f - 2.f * u + p * q, 1e-15f);
            float Ac = -(1.f - 2.f * u + q) / den;   // coefficient on x_i in m
            float Bc = (1.f - p) / den;              // coefficient on x_j in m
            float mn2 = fmaxf(Ac * Ac * p + 2.f * Ac * Bc * u + Bc * Bc * q, 0.f);
            float mn = sqrtf(mn2);
            float sc = fmaxf(1.f - p, EPSF) * atanhf(fminf(mn, MAXNORM)) / fmaxf(mn, EPSF);
            cacc[r] += w * sc * Ac;
            ldsG[mrow * TILE + l] = w * sc * Bc;     // C layout -> LDS for transpose
        }
        __syncthreads();

        // ---- Stage 2: acc += G (16x16) @ X_j (16x128)  (32x v_wmma f32) ----
#pragma unroll
        for (int kc = 0; kc < 4; ++kc) {
            v2f a = *(const v2f*)(&ldsG[l * TILE + 4 * kc + 2 * g]); // A layout
#pragma unroll
            for (int nc = 0; nc < 8; ++nc) {
                v2f b;
                b.x = cur[(4 * kc + 2 * g) * DD + 16 * nc + l];
                b.y = cur[(4 * kc + 2 * g + 1) * DD + 16 * nc + l];
                acc[nc] = __builtin_amdgcn_wmma_f32_16x16x4_f32(false, a, false, b,
                                                                (short)0, acc[nc],
                                                                false, false);
            }
        }
        __syncthreads();
    }

    // ---- reduce the x_i coefficient across the 16 lanes of each half-wave ----
#pragma unroll
    for (int r = 0; r < 8; ++r) {
#pragma unroll
        for (int m = 8; m >= 1; m >>= 1)
            cacc[r] += __shfl_xor(cacc[r], m, 32);
    }

    // ---- mean_t = cacc * x_i + acc  -> LDS (for A-layout reads) ----
#pragma unroll
    for (int nc = 0; nc < 8; ++nc) {
#pragma unroll
        for (int r = 0; r < 8; ++r) {
            int mrow = r + 8 * g;
            int d = 16 * nc + l;
            float xe = x[(size_t)(i0 + mrow) * DD + d];
            ldsMT[mrow * DD + d] = acc[nc][r] + cacc[r] * xe;
        }
    }
    __syncthreads();

    // ---- T = mean_t (16x128) @ W (128x128) + bias  (256x v_wmma f32) ----
    // Reuse the (now idle) ldsXJ buffers to double-buffer 128x16 panels of W.
    async_copy16(W, &ldsXJ[0], wg, wl);            // panel nc = 0
#pragma unroll
    for (int nc = 0; nc < 8; ++nc) {
        const int buf = nc & 1;
        float* wp = &ldsXJ[buf * (TILE * DD)];
        const bool has_next = nc < 7;
        if (has_next)
            async_copy16(W + (nc + 1) * TILE, &ldsXJ[(buf ^ 1) * (TILE * DD)], wg, wl);
        wait_tile_ready(has_next);

        v8f T = {};
#pragma unroll
        for (int kc = 0; kc < 32; ++kc) {
            v2f a = *(const v2f*)(&ldsMT[l * DD + 4 * kc + 2 * g]);
            v2f b;
            b.x = wp[(4 * kc + 2 * g) * TILE + l];
            b.y = wp[(4 * kc + 2 * g + 1) * TILE + l];
            T = __builtin_amdgcn_wmma_f32_16x16x4_f32(false, a, false, b,
                                                      (short)0, T, false, false);
        }
        float bv = bias[16 * nc + l];
#pragma unroll
        for (int r = 0; r < 8; ++r) T[r] += bv;
        acc[nc] = T;   // reuse accumulator registers for the transformed tile
    }

    // ---- expmap epilogue: row norms + <x, v> via in-register + shfl reduce ----
    float vn2p[8], xvp[8];
#pragma unroll
    for (int r = 0; r < 8; ++r) { vn2p[r] = 0.f; xvp[r] = 0.f; }
#pragma unroll
    for (int nc = 0; nc < 8; ++nc) {
#pragma unroll
        for (int r = 0; r < 8; ++r) {
            int mrow = r + 8 * g;
            float v = acc[nc][r];
            float xe = x[(size_t)(i0 + mrow) * DD + 16 * nc + l];
            vn2p[r] += v * v;
            xvp[r] += xe * v;
        }
    }
#pragma unroll
    for (int r = 0; r < 8; ++r) {
#pragma unroll
        for (int m = 8; m >= 1; m >>= 1) {
            vn2p[r] += __shfl_xor(vn2p[r], m, 32);
            xvp[r]  += __shfl_xor(xvp[r], m, 32);
        }
    }

    float fa[8], fb[8];
#pragma unroll
    for (int r = 0; r < 8; ++r) {
        float p = p_r[r];
        float vn = sqrtf(fmaxf(vn2p[r], 0.f));
        float t = tanhf(vn / fmaxf(1.f - p, EPSF));   // tanh(0.5*lambda*|v|)
        float f = t / fmaxf(vn, EPSF);                // second = f * v
        float y2 = f * f * vn2p[r];
        float xy = f * xvp[r];
        float den = fmaxf(1.f + 2.f * xy + p * y2, 1e-15f);
        fa[r] = (1.f + 2.f * xy + y2) / den;          // coefficient on x_i
        fb[r] = (1.f - p) * f / den;                  // coefficient on v
    }

#pragma unroll
    for (int nc = 0; nc < 8; ++nc) {
#pragma unroll
        for (int r = 0; r < 8; ++r) {
            int mrow = r + 8 * g;
            int d = 16 * nc + l;
            float xe = x[(size_t)(i0 + mrow) * DD + d];
            out[(size_t)(i0 + mrow) * DD + d] = fa[r] * xe + fb[r] * acc[nc][r];
        }
    }
}

// ---------------------------------------------------------------------------
extern "C" void kernel_launch(void* const* d_in, const int* in_sizes, int n_in,
                              void* d_out, int out_size, void* d_ws, size_t ws_size,
                              hipStream_t stream) {
    const float* x    = (const float*)d_in[0];
    const float* adj  = (const float*)d_in[1];
    const float* W    = (const float*)d_in[2];
    const float* bias = (const float*)d_in[3];

    float* px     = (float*)d_ws;        // NN floats
    float* invdeg = px + NN;             // NN floats  (16 KB of ws total)

    precompute_kernel<<<NN, 256, 0, stream>>>(x, adj, px, invdeg);
    manifold_mp_kernel<<<NN / TILE, 32, 0, stream>>>(x, adj, W, bias, px, invdeg,
                                                     (float*)d_out);
}